// RetinaDecoder_19267223290024
// MI455X (gfx1250) — compile-verified
//
#include <hip/hip_runtime.h>
#include <stdint.h>

// ---------------- problem constants ----------------
#define BATCH 4
#define N0 57600
#define N1 14400
#define N2 3600
#define N3 900
#define N4 225
#define NTMP 75600      // levels 0-2 concatenated
#define M 4125          // 1000+1000+1000+900+225 selected per image
#define MP 4128         // padded to 16 multiple
#define WROW 129        // 4128/32 words per suppression row
#define MIN_SCORE 0.05f
#define NMS_THR 0.5f
#define MAX_DET 100

typedef float v2f __attribute__((ext_vector_type(2)));
typedef float v8f __attribute__((ext_vector_type(8)));

// monotonic key: larger float -> larger unsigned
__device__ __forceinline__ unsigned fkey(float f) {
  unsigned u = __float_as_uint(f);
  return (u & 0x80000000u) ? ~u : (u | 0x80000000u);
}

// ---------------- K1: class max/argmax for levels 0-2 ----------------
__global__ void k_maxcls012(const float* __restrict__ c0, const float* __restrict__ c1,
                            const float* __restrict__ c2,
                            float* __restrict__ scores_tmp, int* __restrict__ cls_tmp) {
  int gid = blockIdx.x * blockDim.x + threadIdx.x;
  const int NT = BATCH * NTMP;
  if (gid >= NT) return;
  int b = gid / NTMP;
  int r = gid - b * NTMP;
  const float* src;
  if (r < N0)            src = c0 + ((size_t)b * N0 + r) * 80;
  else if (r < N0 + N1)  src = c1 + ((size_t)b * N1 + (r - N0)) * 80;
  else                   src = c2 + ((size_t)b * N2 + (r - N0 - N1)) * 80;
  __builtin_prefetch(src, 0, 1);
  const float4* s4 = (const float4*)src;
  float best = -__builtin_inff(); int bi = 0;
  #pragma unroll 5
  for (int q = 0; q < 20; ++q) {
    float4 v = s4[q];
    int j = q * 4;
    if (v.x > best) { best = v.x; bi = j; }
    if (v.y > best) { best = v.y; bi = j + 1; }
    if (v.z > best) { best = v.z; bi = j + 2; }
    if (v.w > best) { best = v.w; bi = j + 3; }
  }
  scores_tmp[gid] = best;
  cls_tmp[gid] = bi;
}

// ---------------- K1b: levels 3-4 pass straight into selected arrays ----------------
__global__ void k_maxcls34(const float* __restrict__ c3, const float* __restrict__ c4,
                           const float* __restrict__ r3, const float* __restrict__ r4,
                           const float* __restrict__ a3, const float* __restrict__ a4,
                           float* __restrict__ sel_s, int* __restrict__ sel_c,
                           float* __restrict__ sel_r, float* __restrict__ sel_a) {
  int gid = blockIdx.x * blockDim.x + threadIdx.x;
  const int NT = BATCH * (N3 + N4);
  if (gid >= NT) return;
  int b = gid / (N3 + N4);
  int r = gid - b * (N3 + N4);
  const float* src; const float* rg; const float* an; int dst;
  if (r < N3) {
    src = c3 + ((size_t)b * N3 + r) * 80;
    rg  = r3 + ((size_t)b * N3 + r) * 4;
    an  = a3 + ((size_t)b * N3 + r) * 4;
    dst = b * M + 3000 + r;
  } else {
    int i = r - N3;
    src = c4 + ((size_t)b * N4 + i) * 80;
    rg  = r4 + ((size_t)b * N4 + i) * 4;
    an  = a4 + ((size_t)b * N4 + i) * 4;
    dst = b * M + 3900 + i;
  }
  const float4* s4 = (const float4*)src;
  float best = -__builtin_inff(); int bi = 0;
  for (int q = 0; q < 20; ++q) {
    float4 v = s4[q];
    int j = q * 4;
    if (v.x > best) { best = v.x; bi = j; }
    if (v.y > best) { best = v.y; bi = j + 1; }
    if (v.z > best) { best = v.z; bi = j + 2; }
    if (v.w > best) { best = v.w; bi = j + 3; }
  }
  sel_s[dst] = best;
  sel_c[dst] = bi;
  for (int q = 0; q < 4; ++q) { sel_r[dst * 4 + q] = rg[q]; sel_a[dst * 4 + q] = an[q]; }
}

// ---------------- K2: exact top-1000 per (image, level 0..2) via byte radix-select ----------------
__global__ void __launch_bounds__(256)
k_select(const float* __restrict__ scores_tmp, const int* __restrict__ cls_tmp,
         const float* __restrict__ rg0, const float* __restrict__ rg1, const float* __restrict__ rg2,
         const float* __restrict__ an0, const float* __restrict__ an1, const float* __restrict__ an2,
         float* __restrict__ sel_s, int* __restrict__ sel_c,
         float* __restrict__ sel_r, float* __restrict__ sel_a) {
  __shared__ unsigned hist[256];
  __shared__ unsigned sh_p, sh_t, sh_gt, sh_eq;
  int blk = blockIdx.x;
  int b = blk / 3, lv = blk % 3;
  int N, roff, soff; const float* rg; const float* an;
  if (lv == 0)      { N = N0; roff = 0;        soff = 0;    rg = rg0; an = an0; }
  else if (lv == 1) { N = N1; roff = N0;       soff = 1000; rg = rg1; an = an1; }
  else              { N = N2; roff = N0 + N1;  soff = 2000; rg = rg2; an = an2; }
  const float* sc = scores_tmp + (size_t)b * NTMP + roff;
  unsigned prefix = 0, target = 1000;
  for (int p = 0; p < 4; ++p) {
    hist[threadIdx.x] = 0;
    __syncthreads();
    int shift = 24 - 8 * p;
    for (int e = threadIdx.x; e < N; e += 256) {
      unsigned u = fkey(sc[e]);
      bool ok = (p == 0) || ((u >> (shift + 8)) == prefix);
      if (ok) atomicAdd(&hist[(u >> shift) & 255u], 1u);
    }
    __syncthreads();
    if (threadIdx.x == 0) {
      unsigned cum = 0, chosen = 0;
      for (int bin = 255; bin >= 0; --bin) {
        unsigned c = hist[bin];
        if (cum + c >= target) { chosen = (unsigned)bin; break; }
        cum += c;
      }
      sh_p = (prefix << 8) | chosen;
      sh_t = target - cum;
    }
    __syncthreads();
    prefix = sh_p; target = sh_t;
    __syncthreads();
  }
  unsigned T = prefix;  // full 32-bit threshold key; take all > T, plus `target` equal to T
  if (threadIdx.x == 0) { sh_gt = 0; sh_eq = 1000u - target; }
  __syncthreads();
  for (int e = threadIdx.x; e < N; e += 256) {
    float f = sc[e];
    unsigned u = fkey(f);
    int slot = -1;
    if (u > T) slot = (int)atomicAdd(&sh_gt, 1u);
    else if (u == T) {
      unsigned s2 = atomicAdd(&sh_eq, 1u);
      if (s2 < 1000u) slot = (int)s2;
    }
    if (slot >= 0) {
      int dst = b * M + soff + slot;
      sel_s[dst] = f;
      sel_c[dst] = cls_tmp[(size_t)b * NTMP + roff + e];
      size_t srcb = ((size_t)b * N + e) * 4;
      for (int q = 0; q < 4; ++q) {
        sel_r[dst * 4 + q] = rg[srcb + q];
        sel_a[dst * 4 + q] = an[srcb + q];
      }
    }
  }
}

// ---------------- K3: per-image bitonic sort (desc, stable) + box decode ----------------
__global__ void __launch_bounds__(1024)
k_sortdecode(const float* __restrict__ sel_s, const int* __restrict__ sel_c,
             const float* __restrict__ sel_r, const float* __restrict__ sel_a,
             float* __restrict__ sorted_s, int* __restrict__ sorted_c,
             float* __restrict__ sorted_box, float* __restrict__ sorted_area) {
  __shared__ unsigned long long keys[8192];   // 64 KB
  int b = blockIdx.x, tid = threadIdx.x;
  for (int i = tid; i < 8192; i += 1024) {
    unsigned long long k;
    if (i < M) {
      float f = sel_s[b * M + i];
      float m = (f > MIN_SCORE) ? f : -__builtin_inff();
      unsigned u = fkey(m);
      k = ((unsigned long long)(~u) << 32) | (unsigned)i;  // ascending = score desc, idx asc
    } else {
      k = ~0ull;
    }
    keys[i] = k;
  }
  for (unsigned size = 2; size <= 8192; size <<= 1)
    for (unsigned stride = size >> 1; stride > 0; stride >>= 1) {
      __syncthreads();
      for (unsigned i = tid; i < 8192; i += 1024) {
        unsigned j = i ^ stride;
        if (j > i) {
          bool up = ((i & size) == 0);
          unsigned long long a = keys[i], c = keys[j];
          if ((a > c) == up) { keys[i] = c; keys[j] = a; }
        }
      }
    }
  __syncthreads();
  for (int i = tid; i < MP; i += 1024) {
    int o = b * MP + i;
    if (i < M) {
      int idx = (int)(unsigned)(keys[i] & 0xffffffffull);
      float f = sel_s[b * M + idx];
      float m = (f > MIN_SCORE) ? f : -__builtin_inff();
      sorted_s[o] = m;
      sorted_c[o] = sel_c[b * M + idx];
      const float* an = sel_a + (size_t)(b * M + idx) * 4;
      const float* rg = sel_r + (size_t)(b * M + idx) * 4;
      float w = an[2] - an[0], h = an[3] - an[1];
      float cx = an[0] + 0.5f * w, cy = an[1] + 0.5f * h;
      float tx = rg[0] * 0.1f, ty = rg[1] * 0.1f;
      float tw = rg[2] * 0.2f, th = rg[3] * 0.2f;
      float pw = expf(tw) * w, ph = expf(th) * h;
      float pcx = tx * w + cx, pcy = ty * h + cy;
      int x1 = (int)(pcx - 0.5f * pw), y1 = (int)(pcy - 0.5f * ph);
      int x2 = (int)(pcx + 0.5f * pw), y2 = (int)(pcy + 0.5f * ph);
      x1 = x1 > 0 ? x1 : 0;  y1 = y1 > 0 ? y1 : 0;
      x2 = x2 < 639 ? x2 : 639;  y2 = y2 < 639 ? y2 : 639;
      float fx1 = (float)x1, fy1 = (float)y1, fx2 = (float)x2, fy2 = (float)y2;
      sorted_box[o * 4 + 0] = fx1; sorted_box[o * 4 + 1] = fy1;
      sorted_box[o * 4 + 2] = fx2; sorted_box[o * 4 + 3] = fy2;
      sorted_area[o] = (fx2 - fx1) * (fy2 - fy1);
    } else {
      sorted_s[o] = -__builtin_inff();
      sorted_c[o] = 0;
      sorted_box[o * 4 + 0] = 0.f; sorted_box[o * 4 + 1] = 0.f;
      sorted_box[o * 4 + 2] = 0.f; sorted_box[o * 4 + 3] = 0.f;
      sorted_area[o] = 0.f;
    }
  }
}

// ---------------- K4: suppression bit-matrix; area-sum via V_WMMA_F32_16X16X4_F32 ----------------
// S[m][n] = area_i[m] + area_j[n] computed as rank-2 product:
//   A(16x4) = [area_i, 1, 0, 0] per row; B(4x16) rows = [1; area_j; 0; 0]
__global__ void __launch_bounds__(256)
k_ioumat(const float* __restrict__ sorted_box, const float* __restrict__ sorted_area,
         unsigned* __restrict__ rowbits) {
  int b = blockIdx.y;
  int wv = threadIdx.x >> 5, lane = threadIdx.x & 31;
  const int TR = MP / 16, TC = MP / 32, TOT = TR * TC;  // 258 * 129
  int tile = blockIdx.x * 8 + wv;
  if (tile >= TOT) return;
  int tr = tile / TC, tc = tile % TC;
  int r0 = tr * 16, c0 = tc * 32;
  const float* area = sorted_area + (size_t)b * MP;
  const float4* box = (const float4*)(sorted_box + (size_t)b * MP * 4);

  int lm = lane & 15;
  bool lo = lane < 16;
  // A fragment (16x4 f32): lane<16 -> M=lane, K=0/1 in vgpr0/1; lane>=16 -> K=2/3 (zeros)
  v2f af;  af.x = lo ? area[r0 + lm] : 0.f;       af.y = lo ? 1.f : 0.f;
  // B fragments (4x16 f32): lane<16 -> K=0/1 rows in vgpr0/1; lane>=16 -> K=2/3 (zeros)
  v2f b0f; b0f.x = lo ? 1.f : 0.f;                b0f.y = lo ? area[c0 + lm] : 0.f;
  v2f b1f; b1f.x = lo ? 1.f : 0.f;                b1f.y = lo ? area[c0 + 16 + lm] : 0.f;
  v8f z = {0.f, 0.f, 0.f, 0.f, 0.f, 0.f, 0.f, 0.f};
  v8f S0 = __builtin_amdgcn_wmma_f32_16x16x4_f32(false, af, false, b0f, (short)0, z, false, false);
  v8f S1 = __builtin_amdgcn_wmma_f32_16x16x4_f32(false, af, false, b1f, (short)0, z, false, false);

  int j0 = c0 + lm, j1 = c0 + 16 + lm;
  float4 bj0 = box[j0];
  float4 bj1 = box[j1];
  int hi8 = (lane >> 4) << 3;  // C layout: lane>=16 holds rows M=v+8
  #pragma unroll
  for (int v = 0; v < 8; ++v) {
    int i = r0 + v + hi8;
    float4 bi = box[i];
    float xx1 = fmaxf(bi.x, bj0.x), yy1 = fmaxf(bi.y, bj0.y);
    float xx2 = fminf(bi.z, bj0.z), yy2 = fminf(bi.w, bj0.w);
    float inter = fmaxf(xx2 - xx1, 0.f) * fmaxf(yy2 - yy1, 0.f);
    bool p0 = (inter / (S0[v] - inter + 1e-12f)) > NMS_THR;
    xx1 = fmaxf(bi.x, bj1.x); yy1 = fmaxf(bi.y, bj1.y);
    xx2 = fminf(bi.z, bj1.z); yy2 = fminf(bi.w, bj1.w);
    inter = fmaxf(xx2 - xx1, 0.f) * fmaxf(yy2 - yy1, 0.f);
    bool p1 = (inter / (S1[v] - inter + 1e-12f)) > NMS_THR;
    unsigned long long m0 = __ballot((int)p0);
    unsigned long long m1 = __ballot((int)p1);
    unsigned w_lo = (unsigned)((m0 & 0xffffull) | ((m1 & 0xffffull) << 16));
    unsigned w_hi = (unsigned)(((m0 >> 16) & 0xffffull) | (((m1 >> 16) & 0xffffull) << 16));
    int ilo = r0 + v, ihi = r0 + v + 8;
    if (lane == 0 && ilo < M) rowbits[((size_t)b * MP + ilo) * WROW + tc] = w_lo;
    if (lane == 1 && ihi < M) rowbits[((size_t)b * MP + ihi) * WROW + tc] = w_hi;
  }
}

// ---------------- K5: serial greedy NMS over row bitmasks + output ----------------
__global__ void __launch_bounds__(128)
k_nms(const float* __restrict__ sorted_s, const int* __restrict__ sorted_c,
      const float* __restrict__ sorted_box, const unsigned* __restrict__ rowbits,
      float* __restrict__ out) {
  __shared__ unsigned supp[WROW];
  __shared__ int sh_act;
  __shared__ int sh_kept;
  int b = blockIdx.x, tid = threadIdx.x;
  float* out_s = out;                       // [4][100]
  float* out_c = out + BATCH * MAX_DET;     // [4][100]
  float* out_b = out + 2 * BATCH * MAX_DET; // [4][100][4]
  for (int w = tid; w < WROW; w += blockDim.x) supp[w] = 0u;
  for (int k = tid; k < MAX_DET; k += blockDim.x) {
    out_s[b * MAX_DET + k] = -1.f;
    out_c[b * MAX_DET + k] = -1.f;
    for (int q = 0; q < 4; ++q) out_b[(b * MAX_DET + k) * 4 + q] = -1.f;
  }
  if (tid == 0) sh_kept = 0;
  __syncthreads();
  for (int i = 0; i < M; ++i) {
    if (tid == 0) {
      float s = sorted_s[b * MP + i];
      if (!(s > MIN_SCORE)) {
        sh_act = 2;  // sorted desc: nothing valid after this
      } else if ((supp[i >> 5] >> (i & 31)) & 1u) {
        sh_act = 0;
      } else {
        int k = sh_kept;
        out_s[b * MAX_DET + k] = s;
        out_c[b * MAX_DET + k] = (float)sorted_c[b * MP + i];
        for (int q = 0; q < 4; ++q)
          out_b[(b * MAX_DET + k) * 4 + q] = sorted_box[((size_t)b * MP + i) * 4 + q];
        sh_kept = k + 1;
        sh_act = (k + 1 >= MAX_DET) ? 2 : 1;  // first-100 outputs unaffected by later suppression
      }
    }
    __syncthreads();
    int act = sh_act;
    if (act == 2) break;
    if (act == 1) {
      const unsigned* row = rowbits + ((size_t)b * MP + i) * WROW;
      int w0 = i >> 5;
      for (int w = tid; w < WROW; w += blockDim.x) {
        if (w < w0) continue;
        unsigned mword = row[w];
        if (w == w0) mword &= ~((2u << (i & 31)) - 1u);  // keep only bits j > i
        supp[w] |= mword;
      }
    }
    __syncthreads();
  }
}

// ---------------- host side ----------------
extern "C" void kernel_launch(void* const* d_in, const int* in_sizes, int n_in,
                              void* d_out, int out_size, void* d_ws, size_t ws_size,
                              hipStream_t stream) {
  (void)in_sizes; (void)n_in; (void)out_size; (void)ws_size;
  // setup_inputs() dict order: (cls_i, reg_i, anchors_i) per level i
  const float* cls[5]; const float* reg[5]; const float* anc[5];
  for (int i = 0; i < 5; ++i) {
    cls[i] = (const float*)d_in[3 * i + 0];
    reg[i] = (const float*)d_in[3 * i + 1];
    anc[i] = (const float*)d_in[3 * i + 2];
  }
  char* p = (char*)d_ws;
  auto carve = [&](size_t bytes) -> void* {
    void* r = (void*)p;
    p += (bytes + 255) & ~(size_t)255;
    return r;
  };
  float* scores_tmp  = (float*)carve((size_t)BATCH * NTMP * sizeof(float));
  int*   cls_tmp     = (int*)  carve((size_t)BATCH * NTMP * sizeof(int));
  float* sel_s       = (float*)carve((size_t)BATCH * M * sizeof(float));
  int*   sel_c       = (int*)  carve((size_t)BATCH * M * sizeof(int));
  float* sel_r       = (float*)carve((size_t)BATCH * M * 4 * sizeof(float));
  float* sel_a       = (float*)carve((size_t)BATCH * M * 4 * sizeof(float));
  float* sorted_s    = (float*)carve((size_t)BATCH * MP * sizeof(float));
  int*   sorted_c    = (int*)  carve((size_t)BATCH * MP * sizeof(int));
  float* sorted_box  = (float*)carve((size_t)BATCH * MP * 4 * sizeof(float));
  float* sorted_area = (float*)carve((size_t)BATCH * MP * sizeof(float));
  unsigned* rowbits  = (unsigned*)carve((size_t)BATCH * MP * WROW * sizeof(unsigned));

  {
    int nt = BATCH * NTMP;
    k_maxcls012<<<(nt + 255) / 256, 256, 0, stream>>>(cls[0], cls[1], cls[2], scores_tmp, cls_tmp);
  }
  {
    int nt = BATCH * (N3 + N4);
    k_maxcls34<<<(nt + 255) / 256, 256, 0, stream>>>(cls[3], cls[4], reg[3], reg[4], anc[3], anc[4],
                                                     sel_s, sel_c, sel_r, sel_a);
  }
  k_select<<<BATCH * 3, 256, 0, stream>>>(scores_tmp, cls_tmp,
                                          reg[0], reg[1], reg[2],
                                          anc[0], anc[1], anc[2],
                                          sel_s, sel_c, sel_r, sel_a);
  k_sortdecode<<<BATCH, 1024, 0, stream>>>(sel_s, sel_c, sel_r, sel_a,
                                           sorted_s, sorted_c, sorted_box, sorted_area);
  {
    const int TOT = (MP / 16) * (MP / 32);     // 258 * 129 tiles
    dim3 grid((TOT + 7) / 8, BATCH);
    k_ioumat<<<grid, 256, 0, stream>>>(sorted_box, sorted_area, rowbits);
  }
  k_nms<<<BATCH, 128, 0, stream>>>(sorted_s, sorted_c, sorted_box, rowbits, (float*)d_out);
}